// NodeNet_64570538328685
// MI455X (gfx1250) — compile-verified
//
#include <hip/hip_runtime.h>
#include <hip/hip_bf16.h>
#include <math.h>

#define N_NODES 1024
#define N_EDGES 8192
#define NQ 12
#define DIM 4096

typedef float v2f __attribute__((ext_vector_type(2)));
typedef float v8f __attribute__((ext_vector_type(8)));

// ---------------------------------------------------------------------------
// Kernel 0: init M buffer (N x 12): cols 0..7 = 0 (accumulators), cols 8..11 = X
// ---------------------------------------------------------------------------
__global__ void init_M_kernel(const float* __restrict__ X, float* __restrict__ Mbuf) {
    int idx = blockIdx.x * blockDim.x + threadIdx.x;   // over N*12
    if (idx >= N_NODES * 12) return;
    int n = idx / 12, d = idx % 12;
    Mbuf[idx] = (d < 8) ? 0.0f : X[n * 4 + (d - 8)];
}

// ---------------------------------------------------------------------------
// Kernel 1: bo = Ro^T X, bi = Ri^T X  via V_WMMA_F32_16X16X4_F32
// One wave per tile of 16 edges; K = 1024 nodes in steps of 4.
// A[m][k] = R[k][j0+m]  (16x4),  B[k][d] = X[k][d] (4x16, cols 4..15 zero)
// B loads are unconditional (index masked to 0..3) and zeroed by multiplying with a 0/1 mask,
// so no EXEC-divergent regions appear in the K loop.
// ---------------------------------------------------------------------------
__global__ void edge_feat_wmma_kernel(const float* __restrict__ X,
                                      const float* __restrict__ Ri,
                                      const float* __restrict__ Ro,
                                      float* __restrict__ bo,
                                      float* __restrict__ bi) {
    int wave = (blockIdx.x * blockDim.x + threadIdx.x) >> 5;   // 0..1023
    int lane = threadIdx.x & 31;
    const float* R   = (wave < 512) ? Ro : Ri;
    float*       dst = (wave < 512) ? bo : bi;
    int tile = wave & 511;
    int j0 = tile * 16;
    int m  = lane & 15;          // A: row M / B,C: col N
    int K0 = (lane >> 4) * 2;    // lanes 0-15 -> K{0,1}, lanes 16-31 -> K{2,3}
    const float nmask = (m < 4) ? 1.0f : 0.0f;   // zero B cols 4..15
    int mb = m & 3;

    v8f acc = {};
    #pragma unroll 4
    for (int k0 = 0; k0 < N_NODES; k0 += 4) {
        v2f a, b;
        a.x = R[(size_t)(k0 + K0)     * N_EDGES + j0 + m];
        a.y = R[(size_t)(k0 + K0 + 1) * N_EDGES + j0 + m];
        b.x = nmask * X[(k0 + K0)     * 4 + mb];
        b.y = nmask * X[(k0 + K0 + 1) * 4 + mb];
        acc = __builtin_amdgcn_wmma_f32_16x16x4_f32(false, a, false, b,
                                                    (short)0, acc, false, false);
    }
    // C/D layout: lane holds col N=m, rows v + 8*(lane>>4)
    if (m < 4) {
        int rbase = j0 + (lane >> 4) * 8;
        #pragma unroll
        for (int v = 0; v < 8; ++v)
            dst[(rbase + v) * 4 + m] = acc[v];
    }
}

// ---------------------------------------------------------------------------
// Kernel 2: mi = (Ri*e) @ bo -> M[:,0:4], mo = (Ro*e) @ bi -> M[:,4:8]
// One wave per (16-node tile, 512-edge K-chunk); partials via f32 atomics.
// ---------------------------------------------------------------------------
__global__ void aggregate_wmma_kernel(const float* __restrict__ Ri,
                                      const float* __restrict__ Ro,
                                      const float* __restrict__ e,
                                      const float* __restrict__ bo,
                                      const float* __restrict__ bi,
                                      float* __restrict__ Mbuf) {
    int wave = (blockIdx.x * blockDim.x + threadIdx.x) >> 5;   // 0..2047
    int lane = threadIdx.x & 31;
    int mat   = wave >> 10;            // 0: mi (Ri,bo)  1: mo (Ro,bi)
    int rem   = wave & 1023;
    int tile  = rem >> 4;              // 0..63 -> 16 nodes each
    int chunk = rem & 15;              // 0..15 -> 512 edges each
    int n0    = tile * 16;
    int jbase = chunk * 512;
    const float* R = mat ? Ro : Ri;
    const float* B = mat ? bi : bo;
    int m  = lane & 15;
    int K0 = (lane >> 4) * 2;
    const float nmask = (m < 4) ? 1.0f : 0.0f;
    int mb = m & 3;

    v8f acc = {};
    #pragma unroll 4
    for (int j = 0; j < 512; j += 4) {
        int c0 = jbase + j + K0;
        v2f a, b;
        a.x = R[(size_t)(n0 + m) * N_EDGES + c0]     * e[c0];
        a.y = R[(size_t)(n0 + m) * N_EDGES + c0 + 1] * e[c0 + 1];
        b.x = nmask * B[c0 * 4 + mb];
        b.y = nmask * B[(c0 + 1) * 4 + mb];
        acc = __builtin_amdgcn_wmma_f32_16x16x4_f32(false, a, false, b,
                                                    (short)0, acc, false, false);
    }
    if (m < 4) {
        int colbase = mat ? 4 : 0;
        int rbase = n0 + (lane >> 4) * 8;
        #pragma unroll
        for (int v = 0; v < 8; ++v)
            unsafeAtomicAdd(&Mbuf[(rbase + v) * 12 + colbase + m], acc[v]);
    }
}

// ---------------------------------------------------------------------------
// Kernel 3: 12-qubit TTN statevector sim, one block (256 thr) per node.
// State in LDS as split re/im f32. Wire w <-> bit (11-w).
// ---------------------------------------------------------------------------
__device__ __forceinline__ void apply1q(float* re, float* im, int b,
                                        float u00r, float u00i, float u01r, float u01i,
                                        float u10r, float u10i, float u11r, float u11i,
                                        int tid) {
    int mask = (1 << b) - 1;
    #pragma unroll
    for (int k = tid; k < DIM / 2; k += 256) {
        int i0 = ((k >> b) << (b + 1)) | (k & mask);
        int i1 = i0 | (1 << b);
        float a0r = re[i0], a0i = im[i0];
        float a1r = re[i1], a1i = im[i1];
        re[i0] = u00r * a0r - u00i * a0i + u01r * a1r - u01i * a1i;
        im[i0] = u00r * a0i + u00i * a0r + u01r * a1i + u01i * a1r;
        re[i1] = u10r * a0r - u10i * a0i + u11r * a1r - u11i * a1i;
        im[i1] = u10r * a0i + u10i * a0r + u11r * a1i + u11i * a1r;
    }
}

__device__ __forceinline__ void apply_rot(float* re, float* im, int b,
                                          const float* __restrict__ p, int tid) {
    float phi = p[0], th = p[1], om = p[2];
    float c = __cosf(0.5f * th), s = __sinf(0.5f * th);
    float a1 = 0.5f * (phi + om), a2 = 0.5f * (phi - om);
    float c1 = __cosf(a1), s1 = __sinf(a1);
    float c2 = __cosf(a2), s2 = __sinf(a2);
    // u00 = e^{-i a1} c, u01 = -e^{ i a2} s, u10 = e^{-i a2} s, u11 = e^{ i a1} c
    apply1q(re, im, b,
            c1 * c, -s1 * c,
            -c2 * s, -s2 * s,
            c2 * s, -s2 * s,
            c1 * c, s1 * c, tid);
}

__global__ void __launch_bounds__(256)
ttn_circuit_kernel(const float* __restrict__ Mbuf,
                   const float* __restrict__ theta,
                   float* __restrict__ out) {
    __shared__ float re[DIM];
    __shared__ float im[DIM];
    __shared__ float red[256];
    const int tid = threadIdx.x;
    const int n = blockIdx.x;

    for (int i = tid; i < DIM; i += 256) { re[i] = (i == 0) ? 1.0f : 0.0f; im[i] = 0.0f; }
    __syncthreads();

    // initial RY layer
    for (int w = 0; w < NQ; ++w) {
        float ang = Mbuf[n * 12 + w];
        float c = __cosf(0.5f * ang), s = __sinf(0.5f * ang);
        apply1q(re, im, 11 - w, c, 0.f, -s, 0.f, s, 0.f, c, 0.f, tid);
        __syncthreads();
    }

    // TTN blocks: (rot_w1, rot_w2, cnot_c, cnot_t)
    const int BW1[11] = {0, 2, 4, 6, 8, 10, 1, 5, 9, 2, 5};
    const int BW2[11] = {1, 3, 5, 7, 9, 11, 2, 6, 10, 5, 9};
    const int BC[11]  = {0, 3, 4, 7, 8, 11, 1, 6, 10, 2, 5};
    const int BT[11]  = {1, 2, 5, 6, 9, 10, 2, 5, 9, 5, 9};

    int k = 0;
    for (int blk = 0; blk < 11; ++blk) {
        apply_rot(re, im, 11 - BW1[blk], theta + 3 * k, tid); ++k;
        __syncthreads();
        apply_rot(re, im, 11 - BW2[blk], theta + 3 * k, tid); ++k;
        __syncthreads();
        // CNOT: swap target pair where control bit == 1
        int bc = 11 - BC[blk], bt = 11 - BT[blk];
        int lo = bc < bt ? bc : bt, hi = bc < bt ? bt : bc;
        int mlo = (1 << lo) - 1, mhi = (1 << hi) - 1;
        for (int q = tid; q < DIM / 4; q += 256) {
            int t1 = ((q >> lo) << (lo + 1)) | (q & mlo);
            int t2 = ((t1 >> hi) << (hi + 1)) | (t1 & mhi);
            int i0 = t2 | (1 << bc);        // control=1, target=0
            int i1 = i0 | (1 << bt);        // control=1, target=1
            float tr = re[i0]; re[i0] = re[i1]; re[i1] = tr;
            float ti = im[i0]; im[i0] = im[i1]; im[i1] = ti;
        }
        __syncthreads();
    }
    apply_rot(re, im, 11 - 4, theta + 3 * k, tid);   // final Rot on wire 4
    __syncthreads();

    // <Z_9> : sign from bit 2
    float z = 0.0f;
    for (int i = tid; i < DIM; i += 256) {
        float s = ((i >> 2) & 1) ? -1.0f : 1.0f;
        z += s * (re[i] * re[i] + im[i] * im[i]);
    }
    red[tid] = z;
    __syncthreads();
    for (int off = 128; off > 0; off >>= 1) {
        if (tid < off) red[tid] += red[tid + off];
        __syncthreads();
    }
    if (tid == 0) out[n] = 3.14159265358979323846f * (1.0f - red[0]);
}

// ---------------------------------------------------------------------------
extern "C" void kernel_launch(void* const* d_in, const int* in_sizes, int n_in,
                              void* d_out, int out_size, void* d_ws, size_t ws_size,
                              hipStream_t stream) {
    const float* X     = (const float*)d_in[0];   // (1024,4)
    const float* e     = (const float*)d_in[1];   // (8192,)
    const float* Ri    = (const float*)d_in[2];   // (1024,8192)
    const float* Ro    = (const float*)d_in[3];   // (1024,8192)
    const float* theta = (const float*)d_in[4];   // (69,)
    float* out = (float*)d_out;                   // (1024,)

    float* ws   = (float*)d_ws;
    float* bo   = ws;                         // 8192*4
    float* bi   = ws + N_EDGES * 4;           // 8192*4
    float* Mbuf = ws + 2 * N_EDGES * 4;       // 1024*12

    // 0: init M (zeros + X copy)
    init_M_kernel<<<(N_NODES * 12 + 255) / 256, 256, 0, stream>>>(X, Mbuf);
    // 1: bo/bi GEMMs (1024 waves -> 128 blocks of 8 waves)
    edge_feat_wmma_kernel<<<128, 256, 0, stream>>>(X, Ri, Ro, bo, bi);
    // 2: mi/mo GEMMs (2048 waves -> 256 blocks of 8 waves)
    aggregate_wmma_kernel<<<256, 256, 0, stream>>>(Ri, Ro, e, bo, bi, Mbuf);
    // 3: circuit sim, one block per node
    ttn_circuit_kernel<<<N_NODES, 256, 0, stream>>>(Mbuf, theta, out);
}